// MinimalGN_65025804861829
// MI455X (gfx1250) — compile-verified
//
#include <hip/hip_runtime.h>
#include <hip/hip_bf16.h>

typedef __attribute__((ext_vector_type(2))) float v2f;
typedef __attribute__((ext_vector_type(8))) float v8f;

#define DFEAT 128

// ---------------------------------------------------------------------------
// Zero-fill (agg and nodes_agg are adjacent in workspace -> one launch)
// ---------------------------------------------------------------------------
__global__ void gn_zero(float* __restrict__ p, size_t n) {
    size_t i = (size_t)blockIdx.x * blockDim.x + threadIdx.x;
    if (i < n) p[i] = 0.0f;
}

// ---------------------------------------------------------------------------
// Fragment layout for V_WMMA_F32_16X16X4_F32 (wave32):
//   A 16x4 : lanes 0-15 -> M=lane, K={0,1}; lanes 16-31 -> K={2,3}
//   B 4x16 : lane holds col (lane&15), rows half*2 + {0,1}
// With A row-major [M,K] and B = W^T (W row-major [N,K]) both fragments are
// contiguous float2 loads; full unroll turns all addresses into
// base + immediate-offset so the scheduler can pipeline loads ahead of WMMAs.
// ---------------------------------------------------------------------------

// Phase 1: dual GEMM, 16x32 strip per wave (2 col tiles x 2 outputs = 32 acc
// VGPRs -> leaves headroom for load pipelining).
//   out_f = A @ Wf^T + bf   (proj with f_bias baked in)
//   out_g = A @ Wg^T + bg   (node partial, g_bias baked in)
__global__ void gn_gemm_dual(const float* __restrict__ A,
                             const float* __restrict__ Wf,
                             const float* __restrict__ bf,
                             const float* __restrict__ Wg,
                             const float* __restrict__ bg,
                             float* __restrict__ out_f,
                             float* __restrict__ out_g,
                             int M) {
    int wave    = (int)((blockIdx.x * blockDim.x + threadIdx.x) >> 5);
    int nstrips = (M >> 4) * 4;                 // 4 strips of 32 cols each
    if (wave >= nstrips) return;                // wave-uniform -> EXEC all-1s
    int m0   = (wave >> 2) << 4;
    int nb   = (wave & 3) << 5;                 // 0, 32, 64, 96
    int lane = threadIdx.x & 31;
    int half = lane >> 4;
    int l16  = lane & 15;

    const float* Abase  = A  + (size_t)(m0 + l16) * DFEAT + half * 2;
    const float* Wfbase = Wf + (size_t)(nb + l16) * DFEAT + half * 2;
    const float* Wgbase = Wg + (size_t)(nb + l16) * DFEAT + half * 2;

    v8f accf[2] = {};
    v8f accg[2] = {};
#pragma unroll
    for (int k0 = 0; k0 < DFEAT; k0 += 4) {
        v2f a = *(const v2f*)(Abase + k0);
#pragma unroll
        for (int t = 0; t < 2; ++t) {
            v2f bfv = *(const v2f*)(Wfbase + (size_t)t * 16 * DFEAT + k0);
            v2f bgv = *(const v2f*)(Wgbase + (size_t)t * 16 * DFEAT + k0);
            accf[t] = __builtin_amdgcn_wmma_f32_16x16x4_f32(
                false, a, false, bfv, (short)0, accf[t], false, false);
            accg[t] = __builtin_amdgcn_wmma_f32_16x16x4_f32(
                false, a, false, bgv, (short)0, accg[t], false, false);
        }
    }

#pragma unroll
    for (int t = 0; t < 2; ++t) {
        int n0 = nb + t * 16;
        float biasf = bf[n0 + l16];
        float biasg = bg[n0 + l16];
#pragma unroll
        for (int r = 0; r < 8; ++r) {
            int row = m0 + r + half * 8;        // C/D layout: lanes>=16 -> M+8
            size_t idx = (size_t)row * DFEAT + n0 + l16;
            out_f[idx] = accf[t][r] + biasf;
            out_g[idx] = accg[t][r] + biasg;
        }
    }
}

// ---------------------------------------------------------------------------
// Phase 2: edge update + scatter-max.  One wave per edge, each lane owns
// 4 consecutive features (float4 load, coalesced 512B/wave).
// relu(x) >= 0 so u32 atomic-max on the bit pattern == float max; agg is
// zero-initialized so max(agg,0) is automatic AND any zero value is a
// semantic no-op -> skip the atomic entirely (~2x less L2 atomic traffic).
// ---------------------------------------------------------------------------
__global__ void gn_edge(const float* __restrict__ proj_b,
                        const int* __restrict__ senders,
                        const int* __restrict__ receivers,
                        unsigned int* __restrict__ agg_u,
                        int E) {
    int e = (int)((blockIdx.x * blockDim.x + threadIdx.x) >> 5);
    if (e >= E) return;
    int lane = threadIdx.x & 31;
    int s = senders[e];
    int r = receivers[e];
    float4 v = *((const float4*)(proj_b + (size_t)s * DFEAT) + lane);
    unsigned int* dst = agg_u + (size_t)r * DFEAT + lane * 4;
    unsigned int a0 = __float_as_uint(fmaxf(v.x, 0.0f));
    unsigned int a1 = __float_as_uint(fmaxf(v.y, 0.0f));
    unsigned int a2 = __float_as_uint(fmaxf(v.z, 0.0f));
    unsigned int a3 = __float_as_uint(fmaxf(v.w, 0.0f));
    if (a0) atomicMax(dst + 0, a0);
    if (a1) atomicMax(dst + 1, a1);
    if (a2) atomicMax(dst + 2, a2);
    if (a3) atomicMax(dst + 3, a3);
}

// ---------------------------------------------------------------------------
// Phase 3: nodes = node_part(d_out) + agg @ g_in^T ; fused per-graph
// segment-sum of relu(nodes) via atomicAdd (skipping zeros) into
// nodes_agg[G,128].  16x64 strip per wave (4 col tiles, 32 acc VGPRs).
// ---------------------------------------------------------------------------
__global__ void gn_gemm_agg(const float* __restrict__ Agg,
                            const float* __restrict__ Wgin,
                            float* __restrict__ nodes,      // read+write d_out
                            const int* __restrict__ gid,
                            float* __restrict__ nodes_agg,
                            int M) {
    int wave    = (int)((blockIdx.x * blockDim.x + threadIdx.x) >> 5);
    int nstrips = (M >> 4) * 2;                 // 2 strips of 64 cols each
    if (wave >= nstrips) return;
    int m0   = (wave >> 1) << 4;
    int nb   = (wave & 1) << 6;                 // 0 or 64
    int lane = threadIdx.x & 31;
    int half = lane >> 4;
    int l16  = lane & 15;

    const float* Abase = Agg  + (size_t)(m0 + l16) * DFEAT + half * 2;
    const float* Wbase = Wgin + (size_t)(nb + l16) * DFEAT + half * 2;

    v8f acc[4] = {};
#pragma unroll
    for (int k0 = 0; k0 < DFEAT; k0 += 4) {
        v2f a = *(const v2f*)(Abase + k0);
#pragma unroll
        for (int t = 0; t < 4; ++t) {
            v2f b = *(const v2f*)(Wbase + (size_t)t * 16 * DFEAT + k0);
            acc[t] = __builtin_amdgcn_wmma_f32_16x16x4_f32(
                false, a, false, b, (short)0, acc[t], false, false);
        }
    }

#pragma unroll
    for (int r = 0; r < 8; ++r) {
        int row = m0 + r + half * 8;
        int g = gid[row];                       // uniform across 16 lanes
#pragma unroll
        for (int t = 0; t < 4; ++t) {
            int n0 = nb + t * 16;
            size_t idx = (size_t)row * DFEAT + n0 + l16;
            float v = acc[t][r] + nodes[idx];   // node_part already has g_bias
            nodes[idx] = v;                     // pre-relu nodes are the output
            float rv = fmaxf(v, 0.0f);
            if (rv > 0.0f)
                atomicAdd(&nodes_agg[(size_t)g * DFEAT + n0 + l16], rv);
        }
    }
}

// ---------------------------------------------------------------------------
// Phase 4: globals = nodes_agg @ h_nodes^T + h_bias   (tiny: M = 64)
// ---------------------------------------------------------------------------
__global__ void gn_gemm_bias(const float* __restrict__ A,
                             const float* __restrict__ W,
                             const float* __restrict__ bias,
                             float* __restrict__ out,
                             int M) {
    int wave    = (int)((blockIdx.x * blockDim.x + threadIdx.x) >> 5);
    int nstrips = (M >> 4) * 2;
    if (wave >= nstrips) return;
    int m0   = (wave >> 1) << 4;
    int nb   = (wave & 1) << 6;
    int lane = threadIdx.x & 31;
    int half = lane >> 4;
    int l16  = lane & 15;

    const float* Abase = A + (size_t)(m0 + l16) * DFEAT + half * 2;
    const float* Wbase = W + (size_t)(nb + l16) * DFEAT + half * 2;

    v8f acc[4] = {};
#pragma unroll
    for (int k0 = 0; k0 < DFEAT; k0 += 4) {
        v2f a = *(const v2f*)(Abase + k0);
#pragma unroll
        for (int t = 0; t < 4; ++t) {
            v2f b = *(const v2f*)(Wbase + (size_t)t * 16 * DFEAT + k0);
            acc[t] = __builtin_amdgcn_wmma_f32_16x16x4_f32(
                false, a, false, b, (short)0, acc[t], false, false);
        }
    }

#pragma unroll
    for (int t = 0; t < 4; ++t) {
        int n0 = nb + t * 16;
        float bb = bias[n0 + l16];
#pragma unroll
        for (int r = 0; r < 8; ++r) {
            int row = m0 + r + half * 8;
            out[(size_t)row * DFEAT + n0 + l16] = acc[t][r] + bb;
        }
    }
}

// ---------------------------------------------------------------------------
extern "C" void kernel_launch(void* const* d_in, const int* in_sizes, int n_in,
                              void* d_out, int out_size, void* d_ws, size_t ws_size,
                              hipStream_t stream) {
    const float* node_features = (const float*)d_in[0];
    const float* f_sender      = (const float*)d_in[1];
    const float* f_bias        = (const float*)d_in[2];
    const float* g_node        = (const float*)d_in[3];
    const float* g_in          = (const float*)d_in[4];
    const float* g_bias        = (const float*)d_in[5];
    const float* h_nodes       = (const float*)d_in[6];
    const float* h_bias        = (const float*)d_in[7];
    const int*   senders       = (const int*)d_in[8];
    const int*   receivers     = (const int*)d_in[9];
    const int*   graph_ids     = (const int*)d_in[10];

    const int N = in_sizes[0] / DFEAT;
    const int E = in_sizes[8];
    const int G = (out_size - N * DFEAT) / DFEAT;

    // Workspace layout: proj_b [N,128] | agg [N,128] | nodes_agg [G,128]
    float* proj_b    = (float*)d_ws;
    float* agg       = proj_b + (size_t)N * DFEAT;
    float* nodes_agg = agg + (size_t)N * DFEAT;

    float* nodes_out   = (float*)d_out;
    float* globals_out = nodes_out + (size_t)N * DFEAT;

    // Zero agg + nodes_agg (adjacent)
    size_t zn = (size_t)N * DFEAT + (size_t)G * DFEAT;
    gn_zero<<<(unsigned)((zn + 255) / 256), 256, 0, stream>>>(agg, zn);

    // Phase 1: proj_b = nf@f_sender^T + f_bias ; node_part = nf@g_node^T + g_bias
    {
        size_t threads = (size_t)(N / 16) * 4 * 32;
        gn_gemm_dual<<<(unsigned)((threads + 255) / 256), 256, 0, stream>>>(
            node_features, f_sender, f_bias, g_node, g_bias,
            proj_b, nodes_out, N);
    }

    // Phase 2: edge relu + scatter-max
    {
        size_t threads = (size_t)E * 32;
        gn_edge<<<(unsigned)((threads + 255) / 256), 256, 0, stream>>>(
            proj_b, senders, receivers, (unsigned int*)agg, E);
    }

    // Phase 3: nodes += agg@g_in^T ; fused segment-sum of relu(nodes)
    {
        size_t threads = (size_t)(N / 16) * 2 * 32;
        gn_gemm_agg<<<(unsigned)((threads + 255) / 256), 256, 0, stream>>>(
            agg, g_in, nodes_out, graph_ids, nodes_agg, N);
    }

    // Phase 4: globals = nodes_agg @ h_nodes^T + h_bias
    {
        size_t threads = (size_t)(G / 16) * 2 * 32;
        gn_gemm_bias<<<(unsigned)((threads + 255) / 256), 256, 0, stream>>>(
            nodes_agg, h_nodes, h_bias, globals_out, G);
    }
}